// ReconNet_13365938225803
// MI455X (gfx1250) — compile-verified
//
#include <hip/hip_runtime.h>
#include <math.h>

#define NODES_PER_GROUP 10000
#define NEG_INF -1e30f

typedef __attribute__((ext_vector_type(2))) float v2f;
typedef __attribute__((ext_vector_type(8))) float v8f;

// -------------------- edge weights: chunk dots + softmax over C=4 -----------
__global__ void __launch_bounds__(256) ew_kernel(const float* __restrict__ z,
                                                 const int* __restrict__ ei,
                                                 float* __restrict__ ew, int E)
{
    int gid  = blockIdx.x * 256 + threadIdx.x;
    int e    = gid >> 5;
    if (e >= E) return;
    int lane = threadIdx.x & 31;
    int r = ei[e], c = ei[E + e];
    const float4* zr = (const float4*)(z + (size_t)r * 256);
    const float4* zc = (const float4*)(z + (size_t)c * 256);
    float4 a0 = zr[lane * 2 + 0], a1 = zr[lane * 2 + 1];
    float4 b0 = zc[lane * 2 + 0], b1 = zc[lane * 2 + 1];
    float p = a0.x*b0.x + a0.y*b0.y + a0.z*b0.z + a0.w*b0.w
            + a1.x*b1.x + a1.y*b1.y + a1.z*b1.z + a1.w*b1.w;
    // reduce within groups of 8 lanes (one 64-wide chunk per group)
    p += __shfl_xor(p, 1);
    p += __shfl_xor(p, 2);
    p += __shfl_xor(p, 4);
    float s0 = __shfl(p, 0), s1 = __shfl(p, 8), s2 = __shfl(p, 16), s3 = __shfl(p, 24);
    float mx = fmaxf(fmaxf(s0, s1), fmaxf(s2, s3));
    float e0 = __expf(s0 - mx), e1 = __expf(s1 - mx);
    float e2 = __expf(s2 - mx), e3 = __expf(s3 - mx);
    float inv = 1.0f / (e0 + e1 + e2 + e3);
    if (lane == 0) {
        float4 o; o.x = e0*inv; o.y = e1*inv; o.z = e2*inv; o.w = e3*inv;
        *(float4*)(ew + (size_t)e * 4) = o;
    }
}

// -------------------- phi = chunk means of z --------------------------------
__global__ void __launch_bounds__(256) phi_kernel(const float* __restrict__ z,
                                                  float* __restrict__ phi, int N)
{
    int idx = blockIdx.x * 256 + threadIdx.x;
    if (idx >= N * 4) return;
    int n = idx >> 2, c = idx & 3;
    const float* p = z + (size_t)n * 256 + c * 64;
    float s = 0.f;
    #pragma unroll
    for (int i = 0; i < 64; i += 4) {
        float4 v = *(const float4*)(p + i);
        s += v.x + v.y + v.z + v.w;
    }
    phi[idx] = s * (1.0f / 64.0f);
}

// ---------- per (group, column) max & sum-exp over 10000 nodes --------------
__global__ void __launch_bounds__(256) colstat_kernel(const float* __restrict__ phi,
                                                      float* __restrict__ m,
                                                      float* __restrict__ d)
{
    __shared__ float red[256];
    int b = blockIdx.x;
    int g = b >> 2, c = b & 3;
    int base = g * NODES_PER_GROUP;
    int tid = threadIdx.x;
    float mx = NEG_INF;
    for (int i = tid; i < NODES_PER_GROUP; i += 256)
        mx = fmaxf(mx, phi[(size_t)(base + i) * 4 + c]);
    red[tid] = mx; __syncthreads();
    for (int s = 128; s > 0; s >>= 1) {
        if (tid < s) red[tid] = fmaxf(red[tid], red[tid + s]);
        __syncthreads();
    }
    mx = red[0]; __syncthreads();
    float sm = 0.f;
    for (int i = tid; i < NODES_PER_GROUP; i += 256)
        sm += __expf(phi[(size_t)(base + i) * 4 + c] - mx);
    red[tid] = sm; __syncthreads();
    for (int s = 128; s > 0; s >>= 1) {
        if (tid < s) red[tid] += red[tid + s];
        __syncthreads();
    }
    if (tid == 0) { m[b] = mx; d[b] = red[0]; }
}

__global__ void init_s_kernel(float* S) {
    if (threadIdx.x < 2 && blockIdx.x == 0) S[threadIdx.x] = 0.f;
}

// ---------- pc = softmax (in place over phi), s[n], S[g] --------------------
__global__ void __launch_bounds__(256) pc_kernel(float* __restrict__ phi,
                                                 float* __restrict__ s,
                                                 float* __restrict__ S,
                                                 const float* __restrict__ m,
                                                 const float* __restrict__ d, int N)
{
    int n = blockIdx.x * 256 + threadIdx.x;
    if (n >= N) return;
    int g = n / NODES_PER_GROUP;
    float4 p = *(float4*)(phi + (size_t)n * 4);
    float4 pc;
    pc.x = __expf(p.x - m[g * 4 + 0]) / d[g * 4 + 0];
    pc.y = __expf(p.y - m[g * 4 + 1]) / d[g * 4 + 1];
    pc.z = __expf(p.z - m[g * 4 + 2]) / d[g * 4 + 2];
    pc.w = __expf(p.w - m[g * 4 + 3]) / d[g * 4 + 3];
    *(float4*)(phi + (size_t)n * 4) = pc;
    float sn = pc.x + pc.y + pc.z + pc.w;
    s[n] = sn;
    atomicAdd(&S[g], sn);
}

// ---------- nw = pc*(S-s); mask = (nw==rowmax)|(nw>=1) ----------------------
__global__ void __launch_bounds__(256) mask_kernel(const float* __restrict__ pc,
                                                   const float* __restrict__ s,
                                                   const float* __restrict__ S,
                                                   float* __restrict__ maskf, int N)
{
    int n = blockIdx.x * 256 + threadIdx.x;
    if (n >= N) return;
    int g = n / NODES_PER_GROUP;
    float4 p = *(const float4*)(pc + (size_t)n * 4);
    float f = S[g] - s[n];
    float w0 = p.x * f, w1 = p.y * f, w2 = p.z * f, w3 = p.w * f;
    float mx = fmaxf(fmaxf(w0, w1), fmaxf(w2, w3));
    float4 mk;
    mk.x = (w0 == mx || w0 >= 1.0f) ? 1.0f : 0.0f;
    mk.y = (w1 == mx || w1 >= 1.0f) ? 1.0f : 0.0f;
    mk.z = (w2 == mx || w2 >= 1.0f) ? 1.0f : 0.0f;
    mk.w = (w3 == mx || w3 >= 1.0f) ? 1.0f : 0.0f;
    *(float4*)(maskf + (size_t)n * 4) = mk;
}

// ---------- agg0_k = mask_k ? x : 0 (residual init before scatter) ----------
__global__ void __launch_bounds__(256) inita_kernel(const float* __restrict__ x,
                                                    const float* __restrict__ maskf,
                                                    float* __restrict__ agg0, int N)
{
    int idx = blockIdx.x * 256 + threadIdx.x;
    if (idx >= N * 32) return;
    int n = idx >> 5, fg = idx & 31;
    float4 xv = *(const float4*)(x + (size_t)n * 128 + fg * 4);
    float4 mk = *(const float4*)(maskf + (size_t)n * 4);
    float mv[4] = {mk.x, mk.y, mk.z, mk.w};
    #pragma unroll
    for (int k = 0; k < 4; ++k) {
        float4 o;
        o.x = xv.x * mv[k]; o.y = xv.y * mv[k];
        o.z = xv.z * mv[k]; o.w = xv.w * mv[k];
        *(float4*)(agg0 + (size_t)k * N * 128 + (size_t)n * 128 + fg * 4) = o;
    }
}

// ---------- GIN0 scatter, fused over k: agg0_k[col] += x[row]*mask_k*w_k ----
__global__ void __launch_bounds__(256) scatter0_kernel(const float* __restrict__ x,
                                                       const float* __restrict__ maskf,
                                                       const float* __restrict__ ew,
                                                       const int* __restrict__ ei,
                                                       float* __restrict__ agg0,
                                                       int N, int E)
{
    int idx = blockIdx.x * 256 + threadIdx.x;
    if (idx >= E * 32) return;
    int e = idx >> 5, fg = idx & 31;
    int r = ei[e], c = ei[E + e];
    float4 w4 = *(const float4*)(ew + (size_t)e * 4);
    float4 m4 = *(const float4*)(maskf + (size_t)r * 4);
    float4 xv = *(const float4*)(x + (size_t)r * 128 + fg * 4);
    float wk[4] = {w4.x * m4.x, w4.y * m4.y, w4.z * m4.z, w4.w * m4.w};
    #pragma unroll
    for (int k = 0; k < 4; ++k) {
        float coef = wk[k];
        if (coef != 0.0f) {
            float* dst = agg0 + (size_t)k * N * 128 + (size_t)c * 128 + fg * 4;
            atomicAdd(dst + 0, xv.x * coef);
            atomicAdd(dst + 1, xv.y * coef);
            atomicAdd(dst + 2, xv.z * coef);
            atomicAdd(dst + 3, xv.w * coef);
        }
    }
}

// ---------- GIN1 scatter (64 feat): agg[col] += h[row]*w_k ------------------
__global__ void __launch_bounds__(256) scatter1_kernel(const float* __restrict__ h,
                                                       const float* __restrict__ ew,
                                                       const int* __restrict__ ei,
                                                       float* __restrict__ agg,
                                                       int k, int E)
{
    int idx = blockIdx.x * 256 + threadIdx.x;
    if (idx >= E * 16) return;
    int e = idx >> 4, fg = idx & 15;
    int r = ei[e], c = ei[E + e];
    float coef = ew[(size_t)e * 4 + k];
    float4 hv = *(const float4*)(h + (size_t)r * 64 + fg * 4);
    float* dst = agg + (size_t)c * 64 + fg * 4;
    atomicAdd(dst + 0, hv.x * coef);
    atomicAdd(dst + 1, hv.y * coef);
    atomicAdd(dst + 2, hv.z * coef);
    atomicAdd(dst + 3, hv.w * coef);
}

// ---------- decoder scatter (256 feat, unweighted) --------------------------
__global__ void __launch_bounds__(256) scatterd_kernel(const float* __restrict__ h,
                                                       const int* __restrict__ ei,
                                                       float* __restrict__ agg, int E)
{
    int idx = blockIdx.x * 256 + threadIdx.x;
    if (idx >= E * 64) return;
    int e = idx >> 6, fg = idx & 63;
    int r = ei[e], c = ei[E + e];
    float4 hv = *(const float4*)(h + (size_t)r * 256 + fg * 4);
    float* dst = agg + (size_t)c * 256 + fg * 4;
    atomicAdd(dst + 0, hv.x);
    atomicAdd(dst + 1, hv.y);
    atomicAdd(dst + 2, hv.z);
    atomicAdd(dst + 3, hv.w);
}

// ---------- WMMA f32 GEMM: D = epilogue(A@W + bias) -------------------------
// mode 0: +bias; mode 1: relu(+bias); mode 2: relu((+bias)*g*c + b) [BN]
// Each wave computes a 16-row strip x full Nout (NT tiles of 16x16) over K.
__device__ __constant__ float kBnScale = 0.99999500003749969f; // 1/sqrt(1+1e-5)

template<int NT>
__global__ void __launch_bounds__(256) gemm_wmma(const float* __restrict__ A,
                                                 const float* __restrict__ W,
                                                 const float* __restrict__ bias,
                                                 const float* __restrict__ bng,
                                                 const float* __restrict__ bnb,
                                                 float* __restrict__ D,
                                                 float* __restrict__ D2,
                                                 int M, int K, int lda, int ldc,
                                                 int mode)
{
    const int Nout = NT * 16;
    int wave = threadIdx.x >> 5;
    int lane = threadIdx.x & 31;
    int m0 = blockIdx.x * 128 + wave * 16;
    if (m0 >= M) return;
    int hl = lane >> 4;       // which half of the wave
    int mr = lane & 15;       // M index (A/B) / N index (B,C,D)

    v8f acc[NT];
    #pragma unroll
    for (int t = 0; t < NT; ++t)
        #pragma unroll
        for (int j = 0; j < 8; ++j) acc[t][j] = 0.f;

    const float* arow = A + (size_t)(m0 + mr) * lda + 2 * hl;
    for (int k0 = 0; k0 < K; k0 += 4) {
        v2f a;
        a.x = arow[k0];
        a.y = arow[k0 + 1];
        const float* wr = W + (size_t)(k0 + 2 * hl) * Nout + mr;
        #pragma unroll
        for (int t = 0; t < NT; ++t) {
            v2f b;
            b.x = wr[t * 16];
            b.y = wr[Nout + t * 16];
            acc[t] = __builtin_amdgcn_wmma_f32_16x16x4_f32(
                false, a, false, b, (short)0, acc[t], false, false);
        }
    }

    #pragma unroll
    for (int t = 0; t < NT; ++t) {
        int n = t * 16 + mr;
        float bv = bias ? bias[n] : 0.f;
        float sc = 1.f, sh = 0.f;
        if (mode == 2) { sc = bng[n] * kBnScale; sh = bnb[n]; }
        #pragma unroll
        for (int j = 0; j < 8; ++j) {
            float v = acc[t][j] + bv;
            if (mode == 1) v = fmaxf(v, 0.f);
            if (mode == 2) v = fmaxf(v * sc + sh, 0.f);
            size_t idx = (size_t)(m0 + j + 8 * hl) * ldc + n;
            D[idx] = v;
            if (D2) D2[idx] = v;
        }
    }
}

// ============================================================================
extern "C" void kernel_launch(void* const* d_in, const int* in_sizes, int n_in,
                              void* d_out, int out_size, void* d_ws, size_t ws_size,
                              hipStream_t stream)
{
    const float* z       = (const float*)d_in[0];
    const float* x       = (const float*)d_in[1];
    const int*   ei      = (const int*)  d_in[2];
    const float* gin0_W1 = (const float*)d_in[3];
    const float* gin0_b1 = (const float*)d_in[4];
    const float* gin0_W2 = (const float*)d_in[5];
    const float* gin0_b2 = (const float*)d_in[6];
    const float* gin1_W1 = (const float*)d_in[7];
    const float* gin1_b1 = (const float*)d_in[8];
    const float* gin1_W2 = (const float*)d_in[9];
    const float* gin1_b2 = (const float*)d_in[10];
    const float* bn0_g   = (const float*)d_in[11];
    const float* bn0_b   = (const float*)d_in[12];
    const float* bn1_g   = (const float*)d_in[13];
    const float* bn1_b   = (const float*)d_in[14];
    const float* comp_W  = (const float*)d_in[15];
    const float* comp_b  = (const float*)d_in[16];
    const float* dec_W1  = (const float*)d_in[17];
    const float* dec_b1  = (const float*)d_in[18];
    const float* dec_W2  = (const float*)d_in[19];
    const float* dec_b2  = (const float*)d_in[20];
    (void)n_in; (void)out_size; (void)ws_size;

    const int N = in_sizes[1] / 128;
    const int E = in_sizes[2] / 2;

    // ---- workspace layout (floats) ----
    float* ws = (float*)d_ws;
    size_t off = 0;
    auto alloc = [&](size_t nf) { float* p = ws + off; off += (nf + 31) & ~(size_t)31; return p; };
    float* ew    = alloc((size_t)E * 4);
    float* phi   = alloc((size_t)N * 4);        // overwritten with pc
    float* mbuf  = alloc(16);
    float* dbuf  = alloc(16);
    float* Sbuf  = alloc(16);
    float* sbuf  = alloc((size_t)N);
    float* maskf = alloc((size_t)N * 4);
    float* agg0  = alloc((size_t)4 * N * 128);
    float* t64   = alloc((size_t)N * 64);
    float* h1    = alloc((size_t)4 * N * 64);
    float* agg2  = alloc((size_t)4 * N * 64);
    float* hcat  = alloc((size_t)N * 256);
    float* hcomp = alloc((size_t)N * 256);
    float* aggd  = agg0;                        // reuse (agg0 dead by comp time)
    float* t128  = agg0 + (size_t)N * 256;      // reuse second half of agg0

    const int ngroups = N / NODES_PER_GROUP;
    int gemm_blocks = (N + 127) / 128;

    auto gemm = [&](int NT, const float* A, const float* W, const float* b,
                    const float* g, const float* bb, float* D, float* D2,
                    int K, int lda, int ldc, int mode) {
        switch (NT) {
        case 4:  gemm_wmma<4 ><<<gemm_blocks, 256, 0, stream>>>(A, W, b, g, bb, D, D2, N, K, lda, ldc, mode); break;
        case 8:  gemm_wmma<8 ><<<gemm_blocks, 256, 0, stream>>>(A, W, b, g, bb, D, D2, N, K, lda, ldc, mode); break;
        case 16: gemm_wmma<16><<<gemm_blocks, 256, 0, stream>>>(A, W, b, g, bb, D, D2, N, K, lda, ldc, mode); break;
        }
    };

    // 1) edge weights
    ew_kernel<<<(E * 32 + 255) / 256, 256, 0, stream>>>(z, ei, ew, E);
    // 2) node chunk means + group column softmax stats
    phi_kernel<<<(N * 4 + 255) / 256, 256, 0, stream>>>(z, phi, N);
    colstat_kernel<<<ngroups * 4, 256, 0, stream>>>(phi, mbuf, dbuf);
    init_s_kernel<<<1, 32, 0, stream>>>(Sbuf);
    pc_kernel<<<(N + 255) / 256, 256, 0, stream>>>(phi, sbuf, Sbuf, mbuf, dbuf, N);
    mask_kernel<<<(N + 255) / 256, 256, 0, stream>>>(phi, sbuf, Sbuf, maskf, N);

    // 3) GIN layer 0 (all 4 chunks): residual-init, fused scatter, GEMMs
    inita_kernel<<<(N * 32 + 255) / 256, 256, 0, stream>>>(x, maskf, agg0, N);
    scatter0_kernel<<<(E * 32 + 255) / 256, 256, 0, stream>>>(x, maskf, ew, ei, agg0, N, E);
    for (int k = 0; k < 4; ++k) {
        gemm(4, agg0 + (size_t)k * N * 128, gin0_W1, gin0_b1, nullptr, nullptr,
             t64, nullptr, 128, 128, 64, 1);
        // second GEMM + BN-ReLU; dual-store also initializes agg2_k = h1_k
        gemm(4, t64, gin0_W2, gin0_b2, bn0_g, bn0_b,
             h1 + (size_t)k * N * 64, agg2 + (size_t)k * N * 64, 64, 64, 64, 2);
    }

    // 4) GIN layer 1 per chunk; writes directly into concat buffer columns
    for (int k = 0; k < 4; ++k) {
        scatter1_kernel<<<(E * 16 + 255) / 256, 256, 0, stream>>>(
            h1 + (size_t)k * N * 64, ew, ei, agg2 + (size_t)k * N * 64, k, E);
        gemm(4, agg2 + (size_t)k * N * 64, gin1_W1, gin1_b1, nullptr, nullptr,
             t64, nullptr, 64, 64, 64, 1);
        gemm(4, t64, gin1_W2, gin1_b2, bn1_g, bn1_b,
             hcat + k * 64, nullptr, 64, 64, 256, 2);
    }

    // 5) compression GEMM 256x256; dual-store initializes aggd = hcomp
    gemm(16, hcat, comp_W, comp_b, nullptr, nullptr, hcomp, aggd, 256, 256, 256, 0);

    // 6) decoder GIN
    scatterd_kernel<<<(E * 64 + 255) / 256, 256, 0, stream>>>(hcomp, ei, aggd, E);
    gemm(8, aggd, dec_W1, dec_b1, nullptr, nullptr, t128, nullptr, 256, 256, 128, 1);
    gemm(8, t128, dec_W2, dec_b2, nullptr, nullptr, (float*)d_out, nullptr, 128, 128, 128, 0);
}